// Model_15874199126671
// MI455X (gfx1250) — compile-verified
//
#include <hip/hip_runtime.h>
#include <hip/hip_bf16.h>
#include <stdint.h>

// Problem constants from the reference (fixed sizes).
#define N_    2
#define C_    256
#define H_    128
#define W_    128
#define K_    64
#define PH_   16
#define PW_   64
#define HW_   (H_ * W_)            // 16384
#define CHW_  (C_ * HW_)           // 4194304
#define NREG_ (N_ * K_)            // 128
#define SCALE_ 0.25f

// ---- gfx1250 async global->LDS path (probe-guarded) -------------------------
#if defined(__has_builtin)
# if __has_builtin(__builtin_amdgcn_global_load_async_to_lds_b32)
#  define USE_ASYNC_LDS 1
# endif
# if __has_builtin(__builtin_amdgcn_s_wait_asynccnt)
#  define USE_WAIT_BUILTIN 1
# endif
#endif

typedef __attribute__((address_space(1))) int* gint_ptr;
typedef __attribute__((address_space(3))) int* lint_ptr;

__device__ __forceinline__ float bez_eval(float p0, float p1, float p2, float p3, float t) {
    float s = 1.0f - t;
    return p0 * s * s * s + 3.0f * p1 * t * s * s + 3.0f * p2 * t * t * s + p3 * t * t * t;
}

// ---- Pass 1: x = input * masks (masks broadcast over N) ---------------------
__global__ void premul_kernel(const float4* __restrict__ in,
                              const float4* __restrict__ mask,
                              float4* __restrict__ out,
                              int n4, int mask4_mask) {
    int i = blockIdx.x * blockDim.x + threadIdx.x;
    if (i < n4) {
        float4 a = in[i];
        float4 m = mask[i & mask4_mask];   // CHW/4 is a power of two
        out[i] = make_float4(a.x * m.x, a.y * m.y, a.z * m.z, a.w * m.w);
    }
}

// ---- Pass 2: bezier-align gather ------------------------------------------
// Grid: NREG_ * PH_ * 4 blocks; block = 256 threads (8 wave32).
// blockIdx.x -> { rg = b>>6, py = (b>>2)&15, cgroup = b&3 }.
// Each block covers 64 channels x 64 px for one (region, pooled row).
template <bool PRE>
__global__ void __launch_bounds__(256)
bezalign_kernel(const float* __restrict__ src,     // premultiplied (PRE) or raw input
                const float* __restrict__ masks,   // used only when !PRE
                const float* __restrict__ beziers,
                float* __restrict__ out) {
    __shared__ float  lds_cp[16];
    __shared__ int4   lds_off[PW_];
    __shared__ float4 lds_w[PW_];

    const int blk = blockIdx.x;
    const int cg  = blk & 3;
    const int py  = (blk >> 2) & 15;
    const int rg  = blk >> 6;            // 0..127
    const int tid = threadIdx.x;

    // Stage the 16 control points for this region into LDS (async DMA path).
#ifdef USE_ASYNC_LDS
    if (tid < 16) {
        float* gsrc = const_cast<float*>(beziers) + (size_t)rg * 16 + tid;
        __builtin_amdgcn_global_load_async_to_lds_b32(
            (gint_ptr)gsrc,
            (lint_ptr)(&lds_cp[tid]),
            0, 0);
    }
# ifdef USE_WAIT_BUILTIN
    __builtin_amdgcn_s_wait_asynccnt(0);
# else
    asm volatile("s_wait_asynccnt 0x0" ::: "memory");
# endif
#else
    if (tid < 16) lds_cp[tid] = beziers[(size_t)rg * 16 + tid];
#endif
    __syncthreads();

    // Threads 0..63: compute sample coords + bilinear weights for px = tid.
    if (tid < PW_) {
        float cp[16];
#pragma unroll
        for (int i = 0; i < 16; ++i) cp[i] = lds_cp[i] * SCALE_;

        const float u = (float)tid * (1.0f / (float)PW_);
        const float v = (float)py  * (1.0f / (float)PH_);

        // x coords live at even indices, y at odd; first curve = ctrl 0..3.
        float x0 = bez_eval(cp[0],  cp[2],  cp[4],  cp[6],  u);
        float y0 = bez_eval(cp[1],  cp[3],  cp[5],  cp[7],  u);
        float x1 = bez_eval(cp[8],  cp[10], cp[12], cp[14], u);
        float y1 = bez_eval(cp[9],  cp[11], cp[13], cp[15], u);

        float xc = x1 * v + x0 * (1.0f - v) - 0.5f;
        float yc = y1 * v + y0 * (1.0f - v) - 0.5f;

        bool invalid = (yc < -1.0f) || (yc > (float)H_) ||
                       (xc < -1.0f) || (xc > (float)W_);

        float yq = fmaxf(yc, 0.0f);
        float xq = fmaxf(xc, 0.0f);
        int yl = min((int)floorf(yq), H_ - 1);
        int xl = min((int)floorf(xq), W_ - 1);
        int yh = min(yl + 1, H_ - 1);
        int xh = min(xl + 1, W_ - 1);
        float ly = (yl >= H_ - 1) ? 0.0f : (yq - (float)yl);
        float lx = (xl >= W_ - 1) ? 0.0f : (xq - (float)xl);
        float hy = 1.0f - ly, hx = 1.0f - lx;

        float w1 = hy * hx, w2 = hy * lx, w3 = ly * hx, w4 = ly * lx;
        if (invalid) { w1 = w2 = w3 = w4 = 0.0f; }

        lds_off[tid] = make_int4(yl * W_ + xl, yl * W_ + xh,
                                 yh * W_ + xl, yh * W_ + xh);
        lds_w[tid]   = make_float4(w1, w2, w3, w4);
    }
    __syncthreads();

    const int px = tid & 63;
    const int c0 = tid >> 6;                 // 0..3
    const int4   o = lds_off[px];            // ds_load_b128
    const float4 w = lds_w[px];              // ds_load_b128

    const int   n      = rg >> 6;            // batch index (K_ = 64)
    const float* baseN = src + (size_t)n * CHW_;
    // out[((rg*C + c)*PH + py)*PW + px]
    float* obase = out + ((size_t)rg * C_ * PH_ + py) * PW_ + px;

    const int cbeg = cg * 64 + c0;
    const int cend = cg * 64 + 64;

#pragma unroll 4
    for (int c = cbeg; c < cend; c += 4) {
        const float* p = baseN + (size_t)c * HW_;
        float a, b, d, e;
        if (PRE) {
            a = p[o.x]; b = p[o.y]; d = p[o.z]; e = p[o.w];
        } else {
            const float* mp = masks + (size_t)c * HW_;
            a = p[o.x] * mp[o.x];
            b = p[o.y] * mp[o.y];
            d = p[o.z] * mp[o.z];
            e = p[o.w] * mp[o.w];
        }
        obase[(size_t)c * (PH_ * PW_)] =
            fmaf(w.x, a, fmaf(w.y, b, fmaf(w.z, d, w.w * e)));
    }
}

extern "C" void kernel_launch(void* const* d_in, const int* in_sizes, int n_in,
                              void* d_out, int out_size, void* d_ws, size_t ws_size,
                              hipStream_t stream) {
    const float* input   = (const float*)d_in[0];  // (2,256,128,128)
    const float* masks   = (const float*)d_in[1];  // (256,128,128)
    const float* beziers = (const float*)d_in[2];  // (2,64,16)
    float* out = (float*)d_out;                    // (128,256,16,64)

    const size_t need = (size_t)N_ * CHW_ * sizeof(float);   // 33.5 MB
    const dim3 block(256);
    const dim3 grid(NREG_ * PH_ * 4);                        // 8192 blocks

    if (ws_size >= need) {
        float* pm = (float*)d_ws;
        const int n4 = (N_ * CHW_) / 4;                      // 2^21
        premul_kernel<<<dim3((n4 + 255) / 256), block, 0, stream>>>(
            (const float4*)input, (const float4*)masks, (float4*)pm,
            n4, (CHW_ / 4) - 1);
        bezalign_kernel<true><<<grid, block, 0, stream>>>(pm, nullptr, beziers, out);
    } else {
        bezalign_kernel<false><<<grid, block, 0, stream>>>(input, masks, beziers, out);
    }
}